// MaxFlow_67765993996513
// MI455X (gfx1250) — compile-verified
//
#include <hip/hip_runtime.h>
#include <hip/hip_bf16.h>

// ---------------------------------------------------------------------------
// MaxFlow LP via OSQP-style ADMM, restructured with the Woodbury identity:
//   K = a*I_m + rho*A^T A  (m=4096, k=1024)
//   K^-1 r = (1/a) [ r - A^T Ginv (A r) ],  G = (a/rho) I_k + A A^T  (1024^2)
// Setup: At = A^T, G = A@At + diag (WMMA fp32 GEMM, 16x64 strip per wave),
//        Ginv via Newton-Schulz (2 WMMA GEMMs / step, 18 steps).
// Iterations: 400 serial steps, 5 fused kernels each (L2-resident matvecs).
// ---------------------------------------------------------------------------

#define M_EDGES  4096
#define K_NODES  1024
#define N_ITERS  400
#define NS_ITERS 18
#define RHO_C    1.0f
#define SIGMA_C  1e-6f
#define REG_C    1e-6f
#define ALPHA_C  (REG_C + SIGMA_C + RHO_C)   // 1.000002

typedef __attribute__((ext_vector_type(2))) float v2f;
typedef __attribute__((ext_vector_type(8))) float v8f;

#define WMMA_F32(af, bf, acc) \
    __builtin_amdgcn_wmma_f32_16x16x4_f32(false, (af), false, (bf), (short)0, (acc), false, false)

// ------------------------- WMMA fp32 GEMM ----------------------------------
// C[M x N] = alpha*(A@B) + beta*Cin + diag*(row==col).  Row-major.
// Block = 128 threads = 4 waves; block tile = 16 rows x 256 cols.
// Each wave owns a 16x64 strip: 4 accumulators share one (strided) A-fragment
// per K-step, so the uncoalesced A load is amortized 4x; B loads stay
// lane-coalesced. Fragment layouts per CDNA5 ISA 7.12.2:
//   A (16x4): lane L -> M = L%16, K = v + 2*(L>=16)
//   B (4x16): lane L -> N = L%16, K = v + 2*(L>=16)
//   C (16x16): lane L, vgpr v -> M = v + 8*(L>=16), N = L%16
__global__ __launch_bounds__(128)
void gemm_f32_wmma(float* __restrict__ C, const float* __restrict__ A,
                   const float* __restrict__ B, const float* __restrict__ Cin,
                   int M, int N, int K, float alpha, float beta, float diag)
{
    const int lane = threadIdx.x & 31;
    const int wave = threadIdx.x >> 5;          // 0..3
    const int m0   = blockIdx.y * 16;
    const int n0   = blockIdx.x * 256 + wave * 64;
    const int lm   = lane & 15;
    const int hi   = lane >> 4;                 // 0 or 1

    const float* __restrict__ Arow  = A + (size_t)(m0 + lm) * K;
    const float* __restrict__ Bbase = B + (size_t)n0 + lm;

    v8f acc0 = {}, acc1 = {}, acc2 = {}, acc3 = {};
#pragma unroll 4
    for (int kk = 0; kk < K; kk += 4) {
        const int ka = kk + 2 * hi;
        // CDNA5 prefetch path (global_prefetch_b8) a few K-steps ahead
        if ((kk & 63) == 0 && kk + 64 < K) {
            __builtin_prefetch(Arow + kk + 64, 0, 3);
            __builtin_prefetch(Bbase + (size_t)(kk + 64) * N, 0, 3);
        }
        v2f af;
        af.x = Arow[ka];
        af.y = Arow[ka + 1];
        const float* __restrict__ B0 = Bbase + (size_t)ka * N;
        const float* __restrict__ B1 = B0 + N;
        v2f b0, b1, b2, b3;
        b0.x = B0[0];  b0.y = B1[0];
        b1.x = B0[16]; b1.y = B1[16];
        b2.x = B0[32]; b2.y = B1[32];
        b3.x = B0[48]; b3.y = B1[48];
        acc0 = WMMA_F32(af, b0, acc0);
        acc1 = WMMA_F32(af, b1, acc1);
        acc2 = WMMA_F32(af, b2, acc2);
        acc3 = WMMA_F32(af, b3, acc3);
    }

    v8f accs[4] = {acc0, acc1, acc2, acc3};
#pragma unroll
    for (int t = 0; t < 4; ++t) {
        const int col = n0 + t * 16 + lm;
#pragma unroll
        for (int v = 0; v < 8; ++v) {
            const int row = m0 + v + 8 * hi;
            float d = alpha * accs[t][v];
            if (beta != 0.0f) d += beta * Cin[(size_t)row * N + col];
            if (diag != 0.0f && row == col) d += diag;
            C[(size_t)row * N + col] = d;
        }
    }
}

// ------------------------- plain matvec: out = Mat @ v ---------------------
// One wave per row; lanes do float4 coalesced loads; wave32 xor-shuffle reduce.
__global__ __launch_bounds__(256)
void matvec(float* __restrict__ out, const float* __restrict__ Mat,
            const float* __restrict__ v, int rows, int cols)
{
    const int row  = blockIdx.x * 8 + (threadIdx.x >> 5);
    const int lane = threadIdx.x & 31;
    const float* __restrict__ r = Mat + (size_t)row * cols;
    float acc = 0.0f;
    for (int j = lane * 4; j < cols; j += 128) {
        const float4 mm = *(const float4*)(r + j);
        const float4 vv = *(const float4*)(v + j);
        acc += mm.x * vv.x + mm.y * vv.y + mm.z * vv.z + mm.w * vv.w;
    }
#pragma unroll
    for (int off = 16; off > 0; off >>= 1) acc += __shfl_xor(acc, off, 32);
    if (lane == 0) out[row] = acc;
}

// ---- fused rhs: r = At@(rho*z1 - y1) + sigma*x - c + rho*z2 - y2 ----------
__global__ __launch_bounds__(256)
void matvec_rhs(float* __restrict__ r, const float* __restrict__ At,
                const float* __restrict__ z, const float* __restrict__ y,
                const float* __restrict__ x, const float* __restrict__ c,
                int m, int k)
{
    const int row  = blockIdx.x * 8 + (threadIdx.x >> 5);
    const int lane = threadIdx.x & 31;
    const float* __restrict__ Mr = At + (size_t)row * k;
    float acc = 0.0f;
    for (int j = lane * 4; j < k; j += 128) {
        const float4 mm = *(const float4*)(Mr + j);
        const float4 zz = *(const float4*)(z + j);
        const float4 yy = *(const float4*)(y + j);
        acc += mm.x * (RHO_C * zz.x - yy.x) + mm.y * (RHO_C * zz.y - yy.y)
             + mm.z * (RHO_C * zz.z - yy.z) + mm.w * (RHO_C * zz.w - yy.w);
    }
#pragma unroll
    for (int off = 16; off > 0; off >>= 1) acc += __shfl_xor(acc, off, 32);
    if (lane == 0)
        r[row] = acc + SIGMA_C * x[row] - c[row] + RHO_C * z[k + row] - y[k + row];
}

// ---- fused solve+tail-update: x = (r - At@w)/alpha; then tail z/y clip ----
// zt_tail[row] == x_new[row], so the box projection for indices [k, k+m) is
// applied in-register right after the Woodbury completion.
__global__ __launch_bounds__(256)
void matvec_x_update(float* __restrict__ x, float* __restrict__ z,
                     float* __restrict__ y, const float* __restrict__ At,
                     const float* __restrict__ w, const float* __restrict__ r,
                     const float* __restrict__ b, int m, int k)
{
    const int row  = blockIdx.x * 8 + (threadIdx.x >> 5);
    const int lane = threadIdx.x & 31;
    const float* __restrict__ Mr = At + (size_t)row * k;
    float acc = 0.0f;
    for (int j = lane * 4; j < k; j += 128) {
        const float4 mm = *(const float4*)(Mr + j);
        const float4 vv = *(const float4*)(w + j);
        acc += mm.x * vv.x + mm.y * vv.y + mm.z * vv.z + mm.w * vv.w;
    }
#pragma unroll
    for (int off = 16; off > 0; off >>= 1) acc += __shfl_xor(acc, off, 32);
    if (lane == 0) {
        const float xn = (r[row] - acc) * (1.0f / ALPHA_C);
        x[row] = xn;
        const int i = k + row;
        const float zn = fminf(fmaxf(xn + y[i] * (1.0f / RHO_C), 0.0f), b[row]);
        y[i] += RHO_C * (xn - zn);
        z[i] = zn;
    }
}

// ---- fused head-update: t = A@x; z1 = clip(t + y1/rho, 0, 0); y1 update ---
__global__ __launch_bounds__(256)
void matvec_head_update(float* __restrict__ z, float* __restrict__ y,
                        const float* __restrict__ A, const float* __restrict__ x,
                        int k, int m)
{
    const int row  = blockIdx.x * 8 + (threadIdx.x >> 5);
    const int lane = threadIdx.x & 31;
    const float* __restrict__ Mr = A + (size_t)row * m;
    float acc = 0.0f;
    for (int j = lane * 4; j < m; j += 128) {
        const float4 mm = *(const float4*)(Mr + j);
        const float4 vv = *(const float4*)(x + j);
        acc += mm.x * vv.x + mm.y * vv.y + mm.z * vv.z + mm.w * vv.w;
    }
#pragma unroll
    for (int off = 16; off > 0; off >>= 1) acc += __shfl_xor(acc, off, 32);
    if (lane == 0) {
        const float zn = fminf(fmaxf(acc + y[row] * (1.0f / RHO_C), 0.0f), 0.0f);
        y[row] += RHO_C * (acc - zn);
        z[row] = zn;
    }
}

// ------------------------- LDS-tiled transpose -----------------------------
__global__ __launch_bounds__(256)
void transpose_kn(float* __restrict__ At, const float* __restrict__ A,
                  int k, int m)
{
    __shared__ float tile[32][33];
    const int bx = blockIdx.x * 32;   // along m
    const int by = blockIdx.y * 32;   // along k
    const int tx = threadIdx.x & 31;
    const int ty = threadIdx.x >> 5;  // 0..7
#pragma unroll
    for (int i = ty; i < 32; i += 8)
        tile[i][tx] = A[(size_t)(by + i) * m + bx + tx];
    __syncthreads();
#pragma unroll
    for (int i = ty; i < 32; i += 8)
        At[(size_t)(bx + i) * k + by + tx] = tile[tx][i];
}

// --------------- scale = 1 / ||G||_inf (Gershgorin bound) ------------------
__global__ __launch_bounds__(1024)
void ginf_norm(float* __restrict__ scale, const float* __restrict__ G, int k)
{
    __shared__ float red[1024];
    const int t = threadIdx.x;
    float s = 0.0f;
    const float* __restrict__ row = G + (size_t)t * k;
    for (int j = 0; j < k; j += 4) {
        const float4 g = *(const float4*)(row + j);
        s += fabsf(g.x) + fabsf(g.y) + fabsf(g.z) + fabsf(g.w);
    }
    red[t] = s;
    __syncthreads();
    for (int off = 512; off > 0; off >>= 1) {
        if (t < off) red[t] = fmaxf(red[t], red[t + off]);
        __syncthreads();
    }
    if (t == 0) scale[0] = 1.0f / red[0];
}

__global__ __launch_bounds__(256)
void init_X(float* __restrict__ X, const float* __restrict__ scale, int k)
{
    const size_t idx = (size_t)blockIdx.x * blockDim.x + threadIdx.x;
    const int i = (int)(idx / k), j = (int)(idx % k);
    X[idx] = (i == j) ? scale[0] : 0.0f;
}

__global__ __launch_bounds__(256)
void zero_f32(float* __restrict__ p, int n)
{
    const int i = blockIdx.x * blockDim.x + threadIdx.x;
    if (i < n) p[i] = 0.0f;
}

// out[0] = sum_j x[j] * (-c[j])
__global__ __launch_bounds__(1024)
void final_dot(float* __restrict__ out, const float* __restrict__ x,
               const float* __restrict__ c, int m)
{
    __shared__ float red[1024];
    const int t = threadIdx.x;
    float s = 0.0f;
    for (int j = t; j < m; j += 1024) s -= x[j] * c[j];
    red[t] = s;
    __syncthreads();
    for (int off = 512; off > 0; off >>= 1) {
        if (t < off) red[t] += red[t + off];
        __syncthreads();
    }
    if (t == 0) out[0] = red[0];
}

// ---------------------------------------------------------------------------
extern "C" void kernel_launch(void* const* d_in, const int* in_sizes, int n_in,
                              void* d_out, int out_size, void* d_ws, size_t ws_size,
                              hipStream_t stream)
{
    (void)in_sizes; (void)n_in; (void)out_size; (void)ws_size;
    const int m = M_EDGES, k = K_NODES;

    const float* A = (const float*)d_in[0];   // k x m row-major
    const float* b = (const float*)d_in[1];   // m
    const float* c = (const float*)d_in[2];   // m
    float* out = (float*)d_out;

    // Workspace layout (floats), ~34 MB
    float* ws    = (float*)d_ws;
    float* At    = ws;                        // m*k
    float* G     = At + (size_t)m * k;        // k*k
    float* Xa    = G  + (size_t)k * k;        // k*k  (Newton-Schulz ping)
    float* Xb    = Xa + (size_t)k * k;        // k*k  (Newton-Schulz pong)
    float* T     = Xb + (size_t)k * k;        // k*k  (NS temp)
    float* x     = T  + (size_t)k * k;        // m
    float* z     = x  + m;                    // k+m
    float* y     = z  + (k + m);              // k+m
    float* r     = y  + (k + m);              // m
    float* t1    = r  + m;                    // k
    float* w     = t1 + k;                    // k
    float* scale = w  + k;                    // 1

    // ---- Setup: At = A^T -------------------------------------------------
    transpose_kn<<<dim3(m / 32, k / 32), 256, 0, stream>>>(At, A, k, m);

    // ---- G = A @ At + (alpha/rho) * I  (WMMA fp32) -----------------------
    gemm_f32_wmma<<<dim3(k / 256, k / 16), 128, 0, stream>>>(
        G, A, At, G, k, k, m, 1.0f, 0.0f, ALPHA_C / RHO_C);

    // ---- Ginv via Newton-Schulz: X <- 2X - X(GX) -------------------------
    ginf_norm<<<1, 1024, 0, stream>>>(scale, G, k);
    init_X<<<(int)(((size_t)k * k) / 256), 256, 0, stream>>>(Xa, scale, k);
    for (int it = 0; it < NS_ITERS; ++it) {
        gemm_f32_wmma<<<dim3(k / 256, k / 16), 128, 0, stream>>>(
            T, G, Xa, G, k, k, k, 1.0f, 0.0f, 0.0f);           // T = G@Xa
        gemm_f32_wmma<<<dim3(k / 256, k / 16), 128, 0, stream>>>(
            Xb, Xa, T, Xa, k, k, k, -1.0f, 2.0f, 0.0f);        // Xb = 2Xa - Xa@T
        float* tmp = Xa; Xa = Xb; Xb = tmp;
    }
    // Ginv == Xa

    // ---- Zero ADMM state (d_ws is poisoned; must be deterministic) -------
    const int state_n = m + 2 * (k + m);      // x,z,y contiguous
    zero_f32<<<(state_n + 255) / 256, 256, 0, stream>>>(x, state_n);

    // ---- 400 ADMM iterations: 5 fused launches each ----------------------
    for (int it = 0; it < N_ITERS; ++it) {
        // r = At@(rho*z1 - y1) + sigma*x - c + rho*z2 - y2
        matvec_rhs<<<m / 8, 256, 0, stream>>>(r, At, z, y, x, c, m, k);
        // Woodbury: t1 = A@r ; w = Ginv@t1
        matvec<<<k / 8, 256, 0, stream>>>(t1, A, r, k, m);
        matvec<<<k / 8, 256, 0, stream>>>(w, Xa, t1, k, k);
        // x = (r - At@w)/alpha, fused with tail box projection of z/y
        matvec_x_update<<<m / 8, 256, 0, stream>>>(x, z, y, At, w, r, b, m, k);
        // head: zt = A@x, z1 = clip(.,0,0), y1 += rho*(zt - z1)
        matvec_head_update<<<k / 8, 256, 0, stream>>>(z, y, A, x, k, m);
    }

    // ---- objective: -c^T x ----------------------------------------------
    final_dot<<<1, 1024, 0, stream>>>(out, x, c, m);
}